// UpDownMaxPooling_89593017794762
// MI455X (gfx1250) — compile-verified
//
#include <hip/hip_runtime.h>

// Cumulative (prefix) max along rows of 32 independent 1024x1024 fp32 images.
// Memory-bound: 256 MiB min traffic -> ~11us at 23.3 TB/s on MI455X.
// Pass 1: chunked local scan (16 chunks x 64 rows) with a triple-buffered LDS
//         pipeline fed by ASYNCcnt-tracked global_load_async_to_lds_b128.
// Pass 2: boundary-row fixup (race-free max-reduction of chunk totals).

#define B_      32
#define H_      1024
#define W_      1024
#define CHUNK   64
#define NCHUNK  16              // H_/CHUNK
#define SROWS   8               // rows per pipeline stage
#define NSTAGE  (CHUNK / SROWS) // 8
#define NBUF    3               // LDS buffers (stages in flight)
#define ROWB    (W_ * 4)        // row stride in bytes (== LDS stage row stride)

typedef int v4i __attribute__((ext_vector_type(4)));
typedef __attribute__((address_space(1))) v4i gv4i_t;   // global int4
typedef __attribute__((address_space(3))) v4i lv4i_t;   // LDS int4
typedef __attribute__((address_space(3))) void lvoid_t;

// One async b128 copy; IOFFSET is applied by HW to BOTH the global and LDS
// addresses (CDNA5 ISA), and our global/LDS row strides match,
// so 8 rows share one base-pointer pair. cpol=1 => TH_NT (don't let the
// once-read input stream evict the output working set from L2).
template <int R>
__device__ __forceinline__ void issue_rows(const float* gbase, float* lbase) {
    if constexpr (R < SROWS) {
#if __has_builtin(__builtin_amdgcn_global_load_async_to_lds_b128)
        __builtin_amdgcn_global_load_async_to_lds_b128(
            (gv4i_t*)gbase, (lv4i_t*)lbase, /*offset=*/R * ROWB, /*cpol=*/1);
#else
        unsigned lofs = (unsigned)(unsigned long long)(lvoid_t*)lbase + R * ROWB;
        const float* gp = gbase + (size_t)R * W_;
        asm volatile("global_load_async_to_lds_b128 %0, %1, off"
                     :: "v"(lofs), "v"(gp) : "memory");
#endif
    }
}

__device__ __forceinline__ void issue_stage(const float* gbase, float* lbase) {
    issue_rows<0>(gbase, lbase);
}

template <int N>
__device__ __forceinline__ void wait_asynccnt() {
    asm volatile("" ::: "memory");
#if __has_builtin(__builtin_amdgcn_s_wait_asynccnt)
    __builtin_amdgcn_s_wait_asynccnt(N);
#else
    asm volatile("s_wait_asynccnt %0" :: "n"(N));
#endif
    asm volatile("" ::: "memory");
}

template <int N>
__device__ __forceinline__ void wait_dscnt() {
    asm volatile("" ::: "memory");
#if __has_builtin(__builtin_amdgcn_s_wait_dscnt)
    __builtin_amdgcn_s_wait_dscnt(N);
#else
    asm volatile("s_wait_dscnt %0" :: "n"(N));
#endif
    asm volatile("" ::: "memory");
}

__device__ __forceinline__ float4 max4(float4 a, float4 b) {
    a.x = fmaxf(a.x, b.x);
    a.y = fmaxf(a.y, b.y);
    a.z = fmaxf(a.z, b.z);
    a.w = fmaxf(a.w, b.w);
    return a;
}

// ---------------------------------------------------------------------------
// Pass 1 stage body, compile-time recursion so all wait immediates and async
// offsets are ICEs. At the top of stage S, stages S..min(S+2, NSTAGE-1) are
// outstanding, one op each, so waiting asynccnt <= 8*(min(S+2,NSTAGE-1)-S) does not
// wait for stage S until S == NSTAGE-1, where the bound is 0.
// ---------------------------------------------------------------------------
template <int S>
__device__ __forceinline__ void stage_body(const float* g, float* o,
                                           float (&lds)[NBUF][SROWS][W_],
                                           int t4, float4& acc) {
    if constexpr (S < NSTAGE) {
        constexpr int last_issued = (S + NBUF - 1 < NSTAGE - 1) ? (S + NBUF - 1)
                                                                : (NSTAGE - 1);
        wait_asynccnt<8 * (last_issued - S)>();

        constexpr int buf = S % NBUF;
        float4 v[SROWS];
#pragma unroll
        for (int r = 0; r < SROWS; ++r)
            v[r] = *(const float4*)&lds[buf][r][t4];

        // LDS reads must retire before the async engine may refill this buffer.
        wait_dscnt<0>();

        if constexpr (S + NBUF < NSTAGE)
            issue_stage(g + (size_t)(S + NBUF) * SROWS * W_, &lds[buf][0][t4]);

        // Running prefix max + store (RT hint: keep output hot in 192MB L2 so
        // pass 2 re-reads hit L2, not HBM).
#pragma unroll
        for (int r = 0; r < SROWS; ++r) {
            acc = max4(acc, v[r]);
            *(float4*)(o + (size_t)(S * SROWS + r) * W_) = acc;
        }

        stage_body<S + 1>(g, o, lds, t4, acc);
    }
}

// Block = 256 threads, thread t owns columns [4t, 4t+3]; each thread async-
// prefetches row 0 of each 8-row stage it later reads (own columns only) -> no barriers needed.
__global__ __launch_bounds__(256) void cummax_pass1(const float* __restrict__ x,
                                                    float* __restrict__ y) {
    __shared__ float lds[NBUF][SROWS][W_]; // 96 KB -> 3 blocks/WGP

    const int t4 = threadIdx.x * 4;
    const int c = blockIdx.x;      // chunk 0..15
    const int b = blockIdx.y;      // batch 0..31

    const size_t base = ((size_t)b * H_ + (size_t)c * CHUNK) * W_ + (size_t)t4;
    const float* g = x + base;
    float*       o = y + base;

    // Prologue: load row 0 of all NBUF buffers (3 async ops outstanding, cap is 63).
    issue_stage(g + 0 * SROWS * W_, &lds[0][0][t4]);
    issue_stage(g + 1 * SROWS * W_, &lds[1][0][t4]);
    issue_stage(g + 2 * SROWS * W_, &lds[2][0][t4]);

    float4 acc = make_float4(-__builtin_inff(), -__builtin_inff(),
                             -__builtin_inff(), -__builtin_inff());
    stage_body<0>(g, o, lds, t4, acc);
}

// ---------------------------------------------------------------------------
// Pass 2: fix up chunks 1..15. The exclusive prefix for chunk c is the max of
// boundary rows j*CHUNK-1 (j=1..c) from pass 1. Concurrent fixups may have
// already updated some boundary rows, but any updated value is itself a max of
// a subset of the same chunk totals, so max-reducing all c boundary reads
// yields exactly the exclusive prefix regardless of interleaving.
// ---------------------------------------------------------------------------
__global__ __launch_bounds__(256) void cummax_pass2(float* __restrict__ y) {
    const int t = threadIdx.x;
    const int c = blockIdx.x + 1;  // chunk 1..15
    const int b = blockIdx.y;      // batch

    const size_t col = (size_t)b * ((size_t)H_ * W_) + (size_t)(t * 4);

    float4 p = make_float4(-__builtin_inff(), -__builtin_inff(),
                           -__builtin_inff(), -__builtin_inff());
    for (int j = 1; j <= c; ++j) {
        const float4 v = *(const float4*)(y + col + ((size_t)j * CHUNK - 1) * W_);
        p = max4(p, v);
    }

    float* rowp = y + col + (size_t)c * CHUNK * W_;
#pragma unroll 8
    for (int r = 0; r < CHUNK; ++r) {
        float4 v = *(const float4*)(rowp + (size_t)r * W_);
        v = max4(v, p);
        *(float4*)(rowp + (size_t)r * W_) = v;
    }
}

extern "C" void kernel_launch(void* const* d_in, const int* in_sizes, int n_in,
                              void* d_out, int out_size, void* d_ws, size_t ws_size,
                              hipStream_t stream) {
    (void)in_sizes; (void)n_in; (void)d_ws; (void)ws_size; (void)out_size;

    const float* x = (const float*)d_in[0];
    float*       y = (float*)d_out;

    dim3 blk(256);
    dim3 g1(NCHUNK, B_);
    cummax_pass1<<<g1, blk, 0, stream>>>(x, y);

    dim3 g2(NCHUNK - 1, B_);
    cummax_pass2<<<g2, blk, 0, stream>>>(y);
}